// SAGE_85203561218639
// MI455X (gfx1250) — compile-verified
//
#include <hip/hip_runtime.h>

typedef __attribute__((ext_vector_type(2))) float v2f;
typedef __attribute__((ext_vector_type(8))) float v8f;

#define DIM 128
#define SP  132   // LDS pitch (floats): 132 % 64 = 4 -> conflict-free column reads

// ---------------------------------------------------------------- fill zero
__global__ void sage_fill_zero(float* __restrict__ p, long n) {
    long i = (long)blockIdx.x * blockDim.x + threadIdx.x;
    long stride = (long)gridDim.x * blockDim.x;
    for (; i < n; i += stride) p[i] = 0.0f;
}

// ---------------------------------------------------------------- degree
__global__ void sage_degree(const long long* __restrict__ ei, float* __restrict__ deg, int E) {
    int e = blockIdx.x * blockDim.x + threadIdx.x;
    if (e < E) {
        int dst = (int)ei[(long)E + e];
        atomicAdd(&deg[dst], 1.0f);
    }
}

// ---------------------------------------------------------------- scatter-add x[src] -> agg[dst]
// one wave32 per edge; lane owns 4 consecutive columns (float4 gather, 4 f32 atomics)
__global__ __launch_bounds__(256) void sage_scatter(const float* __restrict__ x,
                                                    const long long* __restrict__ ei,
                                                    float* __restrict__ agg, int E) {
    int e    = blockIdx.x * 8 + (threadIdx.x >> 5);
    int lane = threadIdx.x & 31;
    if (e >= E) return;
    int src = (int)ei[e];
    int dst = (int)ei[(long)E + e];
    const float4 v = *(const float4*)(x + (size_t)src * DIM + lane * 4);
    float* d = agg + (size_t)dst * DIM + lane * 4;
    atomicAdd(d + 0, v.x);
    atomicAdd(d + 1, v.y);
    atomicAdd(d + 2, v.z);
    atomicAdd(d + 3, v.w);
}

// ---------------------------------------------------------------- fused SAGE layer GEMM
// out[m,:] = (agg[m,:]/max(deg,1)) @ Wl + bias + self[m,:] @ Wr   (+ optional ReLU)
// block = 256 threads = 8 waves; block owns 16 rows (M-tile) staged in LDS,
// wave w computes the 16x16 tile at columns [16w, 16w+16) via fp32 WMMA.
__global__ __launch_bounds__(256) void sage_gemm(const float* __restrict__ Aagg,
                                                 const float* __restrict__ Aself,
                                                 const float* __restrict__ deg,
                                                 const float* __restrict__ Wl,
                                                 const float* __restrict__ Wr,
                                                 const float* __restrict__ bias,
                                                 float* __restrict__ out,
                                                 int nrows, int do_relu) {
    __shared__ __align__(16) float sAgg[16 * SP];
    __shared__ __align__(16) float sSelf[16 * SP];

    const int m0  = blockIdx.x * 16;
    const int tid = threadIdx.x;

    // stage 16 rows of agg (pre-scaled by 1/deg) and self into LDS
    for (int i = tid * 4; i < 16 * DIM; i += 256 * 4) {
        int r = i >> 7;          // /128
        int c = i & 127;
        int row = m0 + r;
        if (row >= nrows) row = nrows - 1;
        float sc = 1.0f / fmaxf(deg[row], 1.0f);
        float4 va = *(const float4*)(Aagg + (size_t)row * DIM + c);
        va.x *= sc; va.y *= sc; va.z *= sc; va.w *= sc;
        *(float4*)(&sAgg[r * SP + c]) = va;
        *(float4*)(&sSelf[r * SP + c]) = *(const float4*)(Aself + (size_t)row * DIM + c);
    }
    __syncthreads();

    const int wave = tid >> 5;        // 0..7 -> N-tile
    const int lane = tid & 31;
    const int n0   = wave * 16;
    const int half = lane >> 4;       // 0 or 1
    const int l16  = lane & 15;
    const int k0   = half * 2;        // VGPR0 holds K=k0, VGPR1 holds K=k0+1

    v8f acc = {};

    // agg @ Wl
    #pragma unroll 4
    for (int k = 0; k < DIM; k += 4) {
        v2f a, b;
        a.x = sAgg[l16 * SP + k + k0];
        a.y = sAgg[l16 * SP + k + k0 + 1];
        b.x = Wl[(size_t)(k + k0) * DIM + n0 + l16];
        b.y = Wl[(size_t)(k + k0 + 1) * DIM + n0 + l16];
        acc = __builtin_amdgcn_wmma_f32_16x16x4_f32(false, a, false, b,
                                                    (short)0, acc, false, false);
    }
    // self @ Wr
    #pragma unroll 4
    for (int k = 0; k < DIM; k += 4) {
        v2f a, b;
        a.x = sSelf[l16 * SP + k + k0];
        a.y = sSelf[l16 * SP + k + k0 + 1];
        b.x = Wr[(size_t)(k + k0) * DIM + n0 + l16];
        b.y = Wr[(size_t)(k + k0 + 1) * DIM + n0 + l16];
        acc = __builtin_amdgcn_wmma_f32_16x16x4_f32(false, a, false, b,
                                                    (short)0, acc, false, false);
    }

    const float bv = bias[n0 + l16];
    #pragma unroll
    for (int v = 0; v < 8; v++) {
        int m = m0 + v + half * 8;     // C/D layout: VGPR v holds M = v + 8*(lane/16)
        if (m < nrows) {
            float val = acc[v] + bv;
            if (do_relu) val = fmaxf(val, 0.0f);
            out[(size_t)m * DIM + n0 + l16] = val;
        }
    }
}

// ---------------------------------------------------------------- launch
extern "C" void kernel_launch(void* const* d_in, const int* in_sizes, int n_in,
                              void* d_out, int out_size, void* d_ws, size_t ws_size,
                              hipStream_t stream) {
    const float*     x   = (const float*)d_in[0];
    const long long* ei  = (const long long*)d_in[1];   // int64 [2, E]
    const float*     Wl1 = (const float*)d_in[2];
    const float*     bl1 = (const float*)d_in[3];
    const float*     Wr1 = (const float*)d_in[4];
    const float*     Wl2 = (const float*)d_in[5];
    const float*     bl2 = (const float*)d_in[6];
    const float*     Wr2 = (const float*)d_in[7];
    float*           out = (float*)d_out;

    const int N = in_sizes[0] / DIM;     // 100000
    const int E = in_sizes[1] / 2;       // 1600000

    char* ws = (char*)d_ws;
    float* deg = (float*)ws;
    size_t off1 = (((size_t)N * sizeof(float)) + 255) & ~(size_t)255;
    float* agg = (float*)(ws + off1);
    size_t off2 = off1 + ((((size_t)N * DIM * sizeof(float)) + 255) & ~(size_t)255);
    float* h   = (float*)(ws + off2);

    const long aggN = (long)N * DIM;
    const int  mtiles = (N + 15) / 16;

    // ---- layer 1 ----
    sage_fill_zero<<<1024, 256, 0, stream>>>(deg, N);
    sage_fill_zero<<<4096, 256, 0, stream>>>(agg, aggN);
    sage_degree <<<(E + 255) / 256, 256, 0, stream>>>(ei, deg, E);
    sage_scatter<<<(E + 7) / 8,     256, 0, stream>>>(x, ei, agg, E);
    sage_gemm   <<<mtiles,          256, 0, stream>>>(agg, x, deg, Wl1, Wr1, bl1, h, N, 1);

    // ---- layer 2 ----
    sage_fill_zero<<<4096, 256, 0, stream>>>(agg, aggN);
    sage_scatter<<<(E + 7) / 8,     256, 0, stream>>>(h, ei, agg, E);
    sage_gemm   <<<mtiles,          256, 0, stream>>>(agg, h, deg, Wl2, Wr2, bl2, out, N, 0);
}